// U_FUNC_FieldFormer_6408091206333
// MI455X (gfx1250) — compile-verified
//
#include <hip/hip_runtime.h>
#include <hip/hip_bf16.h>
#include <math.h>

typedef float v2f __attribute__((ext_vector_type(2)));
typedef float v8f __attribute__((ext_vector_type(8)));

// ---------------- problem constants (from reference) ----------------
#define BS_    4096
#define NDX_   18
#define NDC_   4
#define CDIM_  54        // C = 3*NDX
#define KN_    32        // K_NEIGH (sequence length)
#define DM_    64        // D_MODEL
#define DFF_   128
#define NH_    4
#define DH_    16
#define EPS_   1e-5f
#define SAT_   0.3f
#define FB_    5.0f

// ---- ff parameter blob offsets: JAX pytree (sorted dict keys) concat ----
// outer: h1_W,h1_b,h2_W,h2_b,hln_b,hln_g,in_W,in_b,layers[2],log_gammas
// layer: lin1_W,lin1_b,lin2_W,lin2_b,ln1_b,ln1_g,ln2_b,ln2_g,out_W,out_b,qkv_W,qkv_b
enum {
  OFF_H1_W = 0,        // 4096
  OFF_H1_B = 4096,     // 64
  OFF_H2_W = 4160,     // 64
  OFF_H2_B = 4224,     // 1
  OFF_HLN_B = 4225,    // 64
  OFF_HLN_G = 4289,    // 64
  OFF_IN_W  = 4353,    // 256
  OFF_IN_B  = 4609,    // 64
  OFF_LAYERS = 4673,
  L_LIN1_W = 0,        // 8192
  L_LIN1_B = 8192,     // 128
  L_LIN2_W = 8320,     // 8192
  L_LIN2_B = 16512,    // 64
  L_LN1_B  = 16576,    // 64
  L_LN1_G  = 16640,    // 64
  L_LN2_B  = 16704,    // 64
  L_LN2_G  = 16768,    // 64
  L_OUT_W  = 16832,    // 4096
  L_OUT_B  = 20928,    // 64
  L_QKV_W  = 20992,    // 12288
  L_QKV_B  = 33280,    // 192
  LAYER_SZ = 33472,
  OFF_LOG_GAMMAS = OFF_LAYERS + 2 * LAYER_SZ,   // 71617
  FF_TOTAL = OFF_LOG_GAMMAS + 3                 // 71620
};

// ======================================================================
// f32 WMMA tile GEMM: C(MxN) = act( scale * A(MxK) @ B(KxN) + bias )
// A row-major (lda).  B element (k,n) = BT ? B[n*ldb+k] : B[k*ldb+n].
// Tiles distributed over the block's wave32s; call sites are wave-uniform
// so EXEC is all-ones at every v_wmma (CDNA5 requirement).
// Fragment layouts per CDNA5 ISA 7.12.2:
//   A 16x4 : lane(0-15)+half, a.x=A[m][k0+2*half], a.y=A[m][k0+2*half+1]
//   B 4x16 : row K = vgpr + 2*half, col = lane&15
//   C 16x16: row M = vgpr + 8*half, col = lane&15
// ======================================================================
template<int M, int N, int K, bool BT, int ACT>
__device__ __forceinline__ void gemm_wmma(
    const float* __restrict__ A, int lda,
    const float* __restrict__ B, int ldb,
    float* __restrict__ C, int ldc,
    const float* __restrict__ bias, float scale)
{
  const int lane = threadIdx.x & 31;
  const int wave = threadIdx.x >> 5;
  const int nw   = blockDim.x >> 5;
  const int half = lane >> 4;      // 0 or 1
  const int l16  = lane & 15;
  constexpr int NT    = N / 16;
  constexpr int TILES = (M / 16) * NT;
  for (int t = wave; t < TILES; t += nw) {
    const int m0 = (t / NT) * 16;
    const int n0 = (t % NT) * 16;
    v8f acc = {0.f, 0.f, 0.f, 0.f, 0.f, 0.f, 0.f, 0.f};
    const int arow = m0 + l16;
    const int bcol = n0 + l16;
#pragma unroll
    for (int k0 = 0; k0 < K; k0 += 4) {
      const int kb = k0 + 2 * half;
      v2f a, b;
      a.x = A[arow * lda + kb];
      a.y = A[arow * lda + kb + 1];
      if (BT) { b.x = B[bcol * ldb + kb];      b.y = B[bcol * ldb + kb + 1]; }
      else    { b.x = B[kb * ldb + bcol];      b.y = B[(kb + 1) * ldb + bcol]; }
      acc = __builtin_amdgcn_wmma_f32_16x16x4_f32(
                /*neg_a=*/false, a, /*neg_b=*/false, b,
                /*c_mod=*/(short)0, acc, /*reuse_a=*/false, /*reuse_b=*/false);
    }
#pragma unroll
    for (int r = 0; r < 8; ++r) {
      const int row = m0 + r + 8 * half;
      float v = acc[r] * scale;
      if (bias) v += bias[bcol];
      if (ACT == 1) v = fmaxf(v, 0.f);
      C[row * ldc + bcol] = v;
    }
  }
}

// residual add + LayerNorm of one 64-wide row (done by one lane per row)
__device__ __forceinline__ void ln_row(float* __restrict__ h,
                                       const float* __restrict__ r,
                                       const float* __restrict__ g,
                                       const float* __restrict__ b)
{
  float m = 0.f;
#pragma unroll
  for (int i = 0; i < DM_; ++i) { h[i] += r[i]; m += h[i]; }
  m *= (1.0f / DM_);
  float v = 0.f;
#pragma unroll
  for (int i = 0; i < DM_; ++i) { const float d = h[i] - m; v += d * d; }
  v *= (1.0f / DM_);
  const float inv = 1.0f / sqrtf(v + EPS_);
#pragma unroll
  for (int i = 0; i < DM_; ++i) h[i] = (h[i] - m) * inv * g[i] + b[i];
}

// ======================================================================
// Kernel A: grid = 2 blocks (block 0 -> ff1, block 1 -> ff2), 256 threads.
// Each block: (1) redundantly reduces s over the batch, (2) builds the 32
// identical tokens, (3) runs the full 2-layer transformer in LDS with f32
// WMMA, (4) head -> scalar prediction into ws[blockIdx].
// ======================================================================
__global__ __launch_bounds__(256)
void ff_pred_kernel(const float* __restrict__ x, const float* __restrict__ xe,
                    const int* __restrict__ offs,
                    const float* __restrict__ ffA, const float* __restrict__ ffB,
                    float* __restrict__ ws_pred)
{
  __shared__ float sH[KN_ * DM_];        // 2048   hidden state
  __shared__ float sQKV[KN_ * 3 * DM_];  // 6144
  __shared__ float sU[KN_ * DFF_];       // 4096   ATT(4x32x32) / out-proj / FF-hidden
  __shared__ float sO[KN_ * DM_];        // 2048   attn concat / FF out
  __shared__ float sT[KN_ * 4];          // tokens
  __shared__ float sRed[256];
  __shared__ float sHm[DM_];
  __shared__ float sT1[DM_];
  __shared__ float sScal[2];

  const int tid = threadIdx.x;
  const float* ff = (blockIdx.x == 0) ? ffA : ffB;

  // ---- (1) s = mean_b( ||x_b|| + ||xe_b|| ) --------------------------
  float acc = 0.f;
  for (int r = tid; r < BS_; r += 256) {
    float sx = 0.f, se = 0.f;
#pragma unroll
    for (int i = 0; i < NDX_; ++i) {
      const float a = x[r * NDX_ + i];  sx += a * a;
      const float b = xe[r * NDX_ + i]; se += b * b;
    }
    acc += sqrtf(sx) + sqrtf(se);
  }
  sRed[tid] = acc;
  __syncthreads();
  for (int st = 128; st > 0; st >>= 1) {
    if (tid < st) sRed[tid] += sRed[tid + st];
    __syncthreads();
  }
  const float sval = sRed[0] * (1.0f / BS_);

  // ---- (2) tokens: [offs * grid_scale * exp(log_gammas), s] ----------
  if (tid < KN_) {
    const float gs0 = 0.2f, gs1 = 0.2f, gs2 = 0.1f;
    sT[tid * 4 + 0] = (float)offs[tid * 3 + 0] * gs0 * expf(ff[OFF_LOG_GAMMAS + 0]);
    sT[tid * 4 + 1] = (float)offs[tid * 3 + 1] * gs1 * expf(ff[OFF_LOG_GAMMAS + 1]);
    sT[tid * 4 + 2] = (float)offs[tid * 3 + 2] * gs2 * expf(ff[OFF_LOG_GAMMAS + 2]);
    sT[tid * 4 + 3] = sval;
  }
  __syncthreads();

  // ---- (3) input projection: H = T(32x4) @ in_W^T + in_b -------------
  gemm_wmma<KN_, DM_, 4, true, 0>(sT, 4, ff + OFF_IN_W, 4, sH, DM_,
                                  ff + OFF_IN_B, 1.0f);
  __syncthreads();

  for (int l = 0; l < 2; ++l) {
    const float* lp = ff + OFF_LAYERS + l * LAYER_SZ;

    // qkv = H @ qkv_W^T + qkv_b  (32x192)
    gemm_wmma<KN_, 3 * DM_, DM_, true, 0>(sH, DM_, lp + L_QKV_W, DM_,
                                          sQKV, 3 * DM_, lp + L_QKV_B, 1.0f);
    __syncthreads();

    // scores_h = (Q_h @ K_h^T) / sqrt(16)   -> sU as 4 x (32x32)
    for (int hh = 0; hh < NH_; ++hh)
      gemm_wmma<KN_, KN_, DH_, true, 0>(sQKV + hh * DH_, 3 * DM_,
                                        sQKV + DM_ + hh * DH_, 3 * DM_,
                                        sU + hh * KN_ * KN_, KN_, nullptr, 0.25f);
    __syncthreads();

    // softmax over each of the 128 rows
    if (tid < NH_ * KN_) {
      float* row = sU + tid * KN_;
      float mx = row[0];
#pragma unroll
      for (int i = 1; i < KN_; ++i) mx = fmaxf(mx, row[i]);
      float sm = 0.f;
#pragma unroll
      for (int i = 0; i < KN_; ++i) { const float e = expf(row[i] - mx); row[i] = e; sm += e; }
      const float inv = 1.0f / sm;
#pragma unroll
      for (int i = 0; i < KN_; ++i) row[i] *= inv;
    }
    __syncthreads();

    // O_h = att_h @ V_h  -> sO (concat heads)
    for (int hh = 0; hh < NH_; ++hh)
      gemm_wmma<KN_, DH_, KN_, false, 0>(sU + hh * KN_ * KN_, KN_,
                                         sQKV + 2 * DM_ + hh * DH_, 3 * DM_,
                                         sO + hh * DH_, DM_, nullptr, 1.0f);
    __syncthreads();

    // out-proj -> sU (first 2048 floats; ATT data is dead now)
    gemm_wmma<KN_, DM_, DM_, true, 0>(sO, DM_, lp + L_OUT_W, DM_,
                                      sU, DM_, lp + L_OUT_B, 1.0f);
    __syncthreads();

    // h = LN(h + attn_out)
    if (tid < KN_) ln_row(sH + tid * DM_, sU + tid * DM_, lp + L_LN1_G, lp + L_LN1_B);
    __syncthreads();

    // FF hidden = relu(H @ lin1_W^T + b)  (32x128) -> sU
    gemm_wmma<KN_, DFF_, DM_, true, 1>(sH, DM_, lp + L_LIN1_W, DM_,
                                       sU, DFF_, lp + L_LIN1_B, 1.0f);
    __syncthreads();

    // FF out = hidden @ lin2_W^T + b -> sO
    gemm_wmma<KN_, DM_, DFF_, true, 0>(sU, DFF_, lp + L_LIN2_W, DFF_,
                                       sO, DM_, lp + L_LIN2_B, 1.0f);
    __syncthreads();

    // h = LN(h + ff)
    if (tid < KN_) ln_row(sH + tid * DM_, sO + tid * DM_, lp + L_LN2_G, lp + L_LN2_B);
    __syncthreads();
  }

  // ---- (4) head ------------------------------------------------------
  if (tid < DM_) {
    float m = 0.f;
#pragma unroll
    for (int s2 = 0; s2 < KN_; ++s2) m += sH[s2 * DM_ + tid];
    sHm[tid] = m * (1.0f / KN_);
  }
  __syncthreads();
  if (tid == 0) {
    float m = 0.f;
    for (int i = 0; i < DM_; ++i) m += sHm[i];
    m *= (1.0f / DM_);
    float v = 0.f;
    for (int i = 0; i < DM_; ++i) { const float d = sHm[i] - m; v += d * d; }
    v *= (1.0f / DM_);
    sScal[0] = m;
    sScal[1] = 1.0f / sqrtf(v + EPS_);
  }
  __syncthreads();
  if (tid < DM_)
    sHm[tid] = (sHm[tid] - sScal[0]) * sScal[1] * ff[OFF_HLN_G + tid] + ff[OFF_HLN_B + tid];
  __syncthreads();
  if (tid < DM_) {
    float a2 = ff[OFF_H1_B + tid];
#pragma unroll
    for (int j = 0; j < DM_; ++j) a2 += ff[OFF_H1_W + tid * DM_ + j] * sHm[j];
    // exact GELU: x * 0.5 * (1 + erf(x/sqrt(2)))
    sT1[tid] = 0.5f * a2 * (1.0f + erff(a2 * 0.70710678118654752f));
  }
  __syncthreads();
  if (tid == 0) {
    float p = ff[OFF_H2_B];
    for (int j = 0; j < DM_; ++j) p += ff[OFF_H2_W + j] * sT1[j];
    ws_pred[blockIdx.x] = p;
  }
}

// ======================================================================
// Kernel B: per-batch head.  w1 = w1_pred*w1_head_W + w1_head_b (972),
// w2 likewise (216) are built once per block in LDS; then each thread
// handles one batch element:  out = tanh( (w2 @ tanh(w1 @ xe)) * SAT )*5 + uref
// ======================================================================
__global__ __launch_bounds__(256)
void ff_apply_kernel(const float* __restrict__ xe, const float* __restrict__ uref,
                     const float* __restrict__ w1W, const float* __restrict__ w1b,
                     const float* __restrict__ w2W, const float* __restrict__ w2b,
                     const float* __restrict__ ws_pred, float* __restrict__ out)
{
  __shared__ float sW1[CDIM_ * NDX_];  // 972
  __shared__ float sW2[NDC_ * CDIM_];  // 216
  const int tid = threadIdx.x;
  const float w1p = ws_pred[0];
  const float w2p = ws_pred[1];
  for (int i = tid; i < CDIM_ * NDX_; i += 256) sW1[i] = w1p * w1W[i] + w1b[i];
  for (int i = tid; i < NDC_ * CDIM_; i += 256) sW2[i] = w2p * w2W[i] + w2b[i];
  __syncthreads();

  const int b = blockIdx.x * 256 + tid;
  if (b >= BS_) return;

  float v[NDX_];
#pragma unroll
  for (int i = 0; i < NDX_; ++i) v[i] = xe[b * NDX_ + i];

  float u[NDC_] = {0.f, 0.f, 0.f, 0.f};
  for (int c = 0; c < CDIM_; ++c) {
    float t = 0.f;
#pragma unroll
    for (int i = 0; i < NDX_; ++i) t += sW1[c * NDX_ + i] * v[i];
    t = tanhf(t);
#pragma unroll
    for (int m = 0; m < NDC_; ++m) u[m] += sW2[m * CDIM_ + c] * t;
  }
#pragma unroll
  for (int m = 0; m < NDC_; ++m)
    out[b * NDC_ + m] = tanhf(u[m] * SAT_) * FB_ + uref[b * NDC_ + m];
}

// ======================================================================
// Launch.  Primary convention: n_in == 11, each reference argument is one
// flat device buffer; pytree dicts (ff1/ff2) are flattened in JAX
// sorted-key order and concatenated.  Fallback (n_in == 75): every leaf
// is a separate buffer -> pack into d_ws via D2D async copies (the leaf
// order equals the blob order, so packing is sequential copies).
// ======================================================================
static const size_t kLeafSizes[33] = {
  4096, 64, 64, 1, 64, 64, 256, 64,                               // outer (pre-layers)
  8192, 128, 8192, 64, 64, 64, 64, 64, 4096, 64, 12288, 192,      // layer 0
  8192, 128, 8192, 64, 64, 64, 64, 64, 4096, 64, 12288, 192,      // layer 1
  3                                                               // log_gammas
};

extern "C" void kernel_launch(void* const* d_in, const int* in_sizes, int n_in,
                              void* d_out, int out_size, void* d_ws, size_t ws_size,
                              hipStream_t stream)
{
  (void)in_sizes; (void)out_size; (void)ws_size;

  float* ws = (float*)d_ws;           // ws[0], ws[1] = w1_pred, w2_pred

  const float* x    = (const float*)d_in[0];
  const float* xe   = (const float*)d_in[1];
  const float* uref = (const float*)d_in[2];
  // d_in[3] = query_idx: mathematically irrelevant (constant field)
  const int*   offs = (const int*)d_in[4];

  const float *ff1, *ff2, *w1W, *w1b, *w2W, *w2b;

  if (n_in >= 75) {
    // leaves passed individually: pack into ws blobs
    float* blob1 = ws + 16;
    float* blob2 = ws + 16 + ((FF_TOTAL + 7) & ~7);
    size_t off = 0;
    for (int i = 0; i < 33; ++i) {
      hipMemcpyAsync(blob1 + off, d_in[5 + i],  kLeafSizes[i] * sizeof(float),
                     hipMemcpyDeviceToDevice, stream);
      hipMemcpyAsync(blob2 + off, d_in[38 + i], kLeafSizes[i] * sizeof(float),
                     hipMemcpyDeviceToDevice, stream);
      off += kLeafSizes[i];
    }
    ff1 = blob1; ff2 = blob2;
    w1W = (const float*)d_in[71]; w1b = (const float*)d_in[72];
    w2W = (const float*)d_in[73]; w2b = (const float*)d_in[74];
  } else {
    ff1 = (const float*)d_in[5];
    ff2 = (const float*)d_in[6];
    w1W = (const float*)d_in[7]; w1b = (const float*)d_in[8];
    w2W = (const float*)d_in[9]; w2b = (const float*)d_in[10];
  }

  // Stage 1: two blocks, one transformer each (ff1, ff2) -> ws[0], ws[1]
  ff_pred_kernel<<<2, 256, 0, stream>>>(x, xe, offs, ff1, ff2, ws);

  // Stage 2: per-batch head over 4096 elements
  ff_apply_kernel<<<BS_ / 256, 256, 0, stream>>>(xe, uref, w1W, w1b, w2W, w2b,
                                                 ws, (float*)d_out);
}